// DA_conv_5712306504087
// MI455X (gfx1250) — compile-verified
//
#include <hip/hip_runtime.h>
#include <hip/hip_bf16.h>

#define NEG_SLOPE 0.1f

typedef float v2f __attribute__((ext_vector_type(2)));
typedef float v8f __attribute__((ext_vector_type(8)));

__device__ __forceinline__ float leaky(float v) { return v >= 0.f ? v : NEG_SLOPE * v; }

// Async global->LDS copy (CDNA5, tracked by ASYNCcnt). lds_off is a byte
// offset relative to the workgroup's LDS base; g is a per-lane global address.
__device__ __forceinline__ void async_ld_b128(unsigned lds_off, const void* g) {
  asm volatile("global_load_async_to_lds_b128 %0, %1, off"
               :: "v"(lds_off), "v"(g) : "memory");
}
__device__ __forceinline__ void wait_asynccnt0() {
  asm volatile("s_wait_asynccnt 0x0" ::: "memory");
}

// ---------------------------------------------------------------------------
// Kernel 1: tiny per-batch MLPs (negligible work, scalar f32).
//   kern[b, c*9 + ky*3 + kx] = leaky(d @ kW1^T) @ kW2^T        [B, 576]
//   att[b, o]                = sigmoid(leaky(d @ caW1^T) @ caW2^T)  [B, 64]
// ---------------------------------------------------------------------------
__global__ void __launch_bounds__(256)
da_prepare(const float* __restrict__ d,      // [16,64]
           const float* __restrict__ kW1,    // [64,64]
           const float* __restrict__ kW2,    // [576,64]
           const float* __restrict__ caW1,   // [8,64]
           const float* __restrict__ caW2,   // [64,8]
           float* __restrict__ ws_kern,      // [16,576]
           float* __restrict__ ws_att)       // [16,64]
{
  __shared__ float s_hid[64];
  __shared__ float s_a1[8];
  const int b   = blockIdx.x;
  const int tid = threadIdx.x;
  const float* db = d + b * 64;

  if (tid < 64) {
    float acc = 0.f;
    #pragma unroll 8
    for (int j = 0; j < 64; ++j) acc += db[j] * kW1[tid * 64 + j];
    s_hid[tid] = leaky(acc);
  } else if (tid < 72) {
    const int r = tid - 64;
    float acc = 0.f;
    #pragma unroll 8
    for (int j = 0; j < 64; ++j) acc += db[j] * caW1[r * 64 + j];
    s_a1[r] = leaky(acc);
  }
  __syncthreads();

  for (int i = tid; i < 576; i += 256) {
    float acc = 0.f;
    #pragma unroll 8
    for (int c = 0; c < 64; ++c) acc += s_hid[c] * kW2[i * 64 + c];
    ws_kern[b * 576 + i] = acc;
  }
  if (tid < 64) {
    float acc = 0.f;
    #pragma unroll
    for (int r = 0; r < 8; ++r) acc += s_a1[r] * caW2[tid * 8 + r];
    ws_att[b * 64 + tid] = 1.f / (1.f + __expf(-acc));
  }
}

// ---------------------------------------------------------------------------
// Kernel 2: fused depthwise-3x3 + leaky + 1x1 conv (WMMA f32) + bias + x*att.
// Block = (b, h, 64-pixel W tile). 256 threads = 8 wave32.
// LDS (dynamic, ~87.5 KB):
//   s_x  [64][3][69]  stride 207 (odd -> conflict-free channel reads);
//                     center row doubles as the residual-x source.
//   s_t  [64][65]     depthwise result t[c][p]
//   s_cw [64][68]     convW[o][c]; stride 68 -> 4*li banks (conflict-free
//                     A-frag) AND 16B-aligned rows for async B128 staging.
//   s_ab [128]        convB (0..63) and att[b] (64..127)
// GEMM: D[o,p] += convW[o,k] * t[k,p] via V_WMMA_F32_16X16X4_F32,
//   M = cout (A rows), N = pixels (lanes -> coalesced 64B stores), K = cin.
// ---------------------------------------------------------------------------
#define XROW   69
#define XSTR   207          // 3*69
#define TSTR   65
#define CWSTR  68
#define XOFF   0
#define TOFF   (64 * XSTR)              // 13248
#define CWOFF  (TOFF + 64 * TSTR)       // 17408 (multiple of 4 -> 16B aligned)
#define ABOFF  (CWOFF + 64 * CWSTR)     // 21760
#define SMEMF  (ABOFF + 128)            // 21888 floats = 87552 B

__global__ void __launch_bounds__(256)
da_main(const float* __restrict__ x,       // [16,64,128,128]
        const float* __restrict__ convW,   // [64,64]
        const float* __restrict__ convB,   // [64]
        const float* __restrict__ ws_kern, // [16,64,9]
        const float* __restrict__ ws_att,  // [16,64]
        float* __restrict__ out)           // [16,64,128,128]
{
  extern __shared__ float smem[];
  float* s_x  = smem + XOFF;
  float* s_t  = smem + TOFF;
  float* s_cw = smem + CWOFF;
  float* s_ab = smem + ABOFF;

  const int tid = threadIdx.x;
  const int bw  = blockIdx.x & 1;
  const int h   = (blockIdx.x >> 1) & 127;
  const int b   = blockIdx.x >> 8;
  const int w0  = bw * 64;

  // ---- convW -> LDS via async B128 DMA (overlaps with x staging) ----
  // 4096 floats = 1024 float4 chunks; 4 per thread. LDS row stride 68 keeps
  // every (o, 4c) destination 16-byte aligned.
  #pragma unroll
  for (int it = 0; it < 4; ++it) {
    const int q  = tid + 256 * it;        // chunk id
    const int o  = q >> 4;
    const int c4 = (q & 15) * 4;
    const unsigned lds_off = (unsigned)((CWOFF + o * CWSTR + c4) * 4);
    async_ld_b128(lds_off, convW + q * 4);
  }

  // ---- convB + att[b] -> LDS ----
  if (tid < 64)       s_ab[tid] = convB[tid];
  else if (tid < 128) s_ab[tid] = ws_att[b * 64 + (tid - 64)];

  // ---- stage x halo tile: 64 ch x 3 rows x 66 cols, float4 interior ----
  {
    const int tx = tid & 31;                       // lane
    for (int rp = tid >> 5; rp < 192; rp += 8) {   // rp uniform per wave
      const int c  = rp / 3;
      const int r  = rp - c * 3;
      const int hh = h + r - 1;
      const bool hok = (hh >= 0) && (hh < 128);
      float* dst = s_x + c * XSTR + r * XROW;
      const float* src = x + ((size_t)(b * 64 + c) * 128 + hh) * 128;
      if (tx < 17) {                               // 17*4 = 68 >= 66 cols
        const int wstart = w0 - 1 + 4 * tx;        // global col of element 0
        float4 v;
        if (hok && wstart >= 0 && wstart + 3 < 128) {
          v = *(const float4*)(src + wstart);      // b128, 4B-aligned ok
        } else {
          v.x = (hok && wstart + 0 >= 0 && wstart + 0 < 128) ? src[wstart + 0] : 0.f;
          v.y = (hok && wstart + 1 >= 0 && wstart + 1 < 128) ? src[wstart + 1] : 0.f;
          v.z = (hok && wstart + 2 >= 0 && wstart + 2 < 128) ? src[wstart + 2] : 0.f;
          v.w = (hok && wstart + 3 >= 0 && wstart + 3 < 128) ? src[wstart + 3] : 0.f;
        }
        const int cb = 4 * tx;                     // LDS col base (0..64)
        dst[cb + 0] = v.x; dst[cb + 1] = v.y;
        dst[cb + 2] = v.z; dst[cb + 3] = v.w;
      }
    }
  }
  wait_asynccnt0();          // convW DMA complete (per wave, before barrier)
  __syncthreads();

  // ---- depthwise 3x3 with dynamic per-(b,c) kernel, rolling window ----
  {
    const int c  = tid & 63;
    const int pb = (tid >> 6) * 16;                 // 16 pixels per thread
    const float* kw = ws_kern + (b * 64 + c) * 9;
    const float k00 = kw[0], k01 = kw[1], k02 = kw[2];
    const float k10 = kw[3], k11 = kw[4], k12 = kw[5];
    const float k20 = kw[6], k21 = kw[7], k22 = kw[8];
    const float* xr0 = s_x + c * XSTR + pb;
    const float* xr1 = xr0 + XROW;
    const float* xr2 = xr1 + XROW;
    float a0 = xr0[0], a1 = xr0[1];
    float b0 = xr1[0], b1 = xr1[1];
    float c0 = xr2[0], c1 = xr2[1];
    float* tq = s_t + c * TSTR + pb;
    #pragma unroll
    for (int j = 0; j < 16; ++j) {
      const float a2 = xr0[j + 2], b2 = xr1[j + 2], c2 = xr2[j + 2];
      float s = k00 * a0 + k01 * a1 + k02 * a2
              + k10 * b0 + k11 * b1 + k12 * b2
              + k20 * c0 + k21 * c1 + k22 * c2;
      tq[j] = leaky(s);
      a0 = a1; a1 = a2; b0 = b1; b1 = b2; c0 = c1; c1 = c2;
    }
  }
  __syncthreads();

  // ---- 1x1 conv as WMMA f32 16x16x4, fused epilogue ----
  const int wave = tid >> 5;
  const int lane = tid & 31;
  const int li   = lane & 15;
  const int lh   = lane >> 4;

  #pragma unroll
  for (int sub = 0; sub < 2; ++sub) {
    const int tileid = wave * 2 + sub;       // 16 tiles over 8 waves
    const int p0 = (tileid >> 2) * 16;       // pixel tile
    const int m0 = (tileid & 3) * 16;        // cout tile
    v8f acc = {0.f, 0.f, 0.f, 0.f, 0.f, 0.f, 0.f, 0.f};

    // A: convW[o][k]  lane li = row M, (2*lh + v) = K within 4-step
    // B: t[k][p]      lane li = col N
    const float* arow = s_cw + (m0 + li) * CWSTR;
    const float* bcol = s_t + p0 + li;
    #pragma unroll
    for (int k0 = 0; k0 < 64; k0 += 4) {
      const int ka = k0 + 2 * lh;
      v2f av; av.x = arow[ka];              av.y = arow[ka + 1];
      v2f bv; bv.x = bcol[ka * TSTR];       bv.y = bcol[(ka + 1) * TSTR];
      acc = __builtin_amdgcn_wmma_f32_16x16x4_f32(
          /*neg_a=*/false, av, /*neg_b=*/false, bv,
          /*c_mod=*/(short)0, acc, /*reuse_a=*/false, /*reuse_b=*/false);
    }

    // Epilogue: + bias + x*att. Residual x comes from the staged center row
    // (LDS hit instead of a global/L2 re-read). Stores: 16 consecutive w per
    // half-wave -> 64B segments.
    const int wcol   = w0 + p0 + li;
    const int xcol   = p0 + li + 1;          // LDS col of global w (= w0+p0+li)
    #pragma unroll
    for (int r = 0; r < 8; ++r) {
      const int o = m0 + r + lh * 8;         // D row layout: VGPR r, lane half
      const float resid = s_x[o * XSTR + XROW + xcol];
      const size_t idx = ((size_t)(b * 64 + o) * 128 + h) * 128 + wcol;
      out[idx] = acc[r] + s_ab[o] + resid * s_ab[64 + o];
    }
  }
}

// ---------------------------------------------------------------------------
extern "C" void kernel_launch(void* const* d_in, const int* in_sizes, int n_in,
                              void* d_out, int out_size, void* d_ws, size_t ws_size,
                              hipStream_t stream) {
  const float* x     = (const float*)d_in[0];
  const float* d     = (const float*)d_in[1];
  const float* kW1   = (const float*)d_in[2];
  const float* kW2   = (const float*)d_in[3];
  const float* convW = (const float*)d_in[4];
  const float* convB = (const float*)d_in[5];
  const float* caW1  = (const float*)d_in[6];
  const float* caW2  = (const float*)d_in[7];
  float* out = (float*)d_out;

  float* ws_kern = (float*)d_ws;           // 16*576 floats
  float* ws_att  = ws_kern + 16 * 576;     // 16*64 floats

  da_prepare<<<16, 256, 0, stream>>>(d, kW1, kW2, caW1, caW2, ws_kern, ws_att);

  const size_t smem = (size_t)SMEMF * sizeof(float);
  da_main<<<16 * 128 * 2, 256, smem, stream>>>(x, convW, convB, ws_kern, ws_att, out);
}